// glcmLoss_4818953306356
// MI455X (gfx1250) — compile-verified
//
#include <hip/hip_runtime.h>
#include <stdint.h>

// ---------------------------------------------------------------------------
// GLCM contrast loss, MI455X (gfx1250, wave32).
//
// contrast[img][a] = (sum over pixels of (i-j)^2) / (4N)   [exact rearrangement
// of the reference's histogram formulation; bin totals are preserved by clip]
// loss = 0.25 * sum_a |c_og[a] - c_gg[a]|
//
// Workspace layout (needs 8,405,056 bytes):
//   [0, 8388608)           uint8  gray[2][16][512][512]
//   [VM_OFF,  +64)         int32  vm[8][2]   (vmin, vmax per image*angle)
//   [PHI_OFF, +8192)       float  Phi[8][256]  (block partials, high 2^16 part)
//   [PLO_OFF, +8192)       float  Plo[8][256]  (block partials, low 16 bits)
// ---------------------------------------------------------------------------

#define HWPX   (512 * 512)          // 262144 pixels per frame
#define NPX    (16 * HWPX)          // 4,194,304 pixels per image set
#define VM_OFF  (2 * (size_t)NPX)   // 8,388,608
#define PHI_OFF (VM_OFF + 64)
#define PLO_OFF (PHI_OFF + 8 * 256 * sizeof(float))

#define IMAX 0x7fffffff
#define IMIN (-2147483647 - 1)

typedef float v2f __attribute__((ext_vector_type(2)));
typedef float v8f __attribute__((ext_vector_type(8)));

// ---------------------------------------------------------------------------
__global__ void k_init(int* __restrict__ vm) {
    int t = threadIdx.x;
    if (t < 8) { vm[2 * t] = IMAX; vm[2 * t + 1] = IMIN; }
}

// ---------------------------------------------------------------------------
// gray[img][n][y][x] = (uint8) rint( (r+g+b)/3 * 255 )   (round-nearest-even,
// matches jnp.round). float4 reads -> b128 loads; packed u32 stores.
__global__ void k_gray(const float* __restrict__ org,
                       const float* __restrict__ gen,
                       uint8_t* __restrict__ gray) {
    const int img = blockIdx.y;
    const float* __restrict__ src = img ? gen : org;
    const long T = (long)blockIdx.x * 256 + threadIdx.x;   // 0 .. 1048575
    const long P = T * 4;                                  // pixel base
    const int  n   = (int)(P >> 18);                       // batch index
    const int  pix = (int)(P & (HWPX - 1));
    const float* base = src + (long)n * (3 * HWPX) + pix;

    __builtin_prefetch(base + 4096, 0, 0);                 // global_prefetch_b8
    float4 r = *(const float4*)(base);
    float4 g = *(const float4*)(base + HWPX);
    float4 b = *(const float4*)(base + 2 * HWPX);

    uint32_t v0 = (uint32_t)(int)rintf(((r.x + g.x + b.x) * (1.0f / 3.0f)) * 255.0f);
    uint32_t v1 = (uint32_t)(int)rintf(((r.y + g.y + b.y) * (1.0f / 3.0f)) * 255.0f);
    uint32_t v2 = (uint32_t)(int)rintf(((r.z + g.z + b.z) * (1.0f / 3.0f)) * 255.0f);
    uint32_t v3 = (uint32_t)(int)rintf(((r.w + g.w + b.w) * (1.0f / 3.0f)) * 255.0f);

    *(uint32_t*)(gray + (size_t)img * NPX + P) =
        v0 | (v1 << 8) | (v2 << 16) | (v3 << 24);
}

// ---------------------------------------------------------------------------
// Shift set (jnp.roll semantics): shifted[y][x] = gray[(y-dx)&511][(x-dy)&511]
//   a0:(1,0)->(y-1,x)  a1:(1,1)->(y-1,x-1)  a2:(0,1)->(y,x-1)  a3:(-1,1)->(y+1,x-1)
// Each thread owns 64 consecutive pixels of one row (512%64==0), processed in
// dword groups; the x-1 column reuses the group's own bytes + 1 wrap byte.
// ---------------------------------------------------------------------------
__global__ void k_minmax(const uint8_t* __restrict__ gray, int* __restrict__ vm) {
    const int img = blockIdx.y;
    const int T = blockIdx.x * 256 + threadIdx.x;          // 0 .. 65535
    const long P = (long)T * 64;
    const int n  = (int)(P >> 18);
    const int q  = (int)(P & (HWPX - 1));
    const int y  = q >> 9;
    const int x0 = q & 511;
    const uint8_t* fr   = gray + (size_t)img * NPX + (size_t)n * HWPX;
    const uint8_t* rowC = fr + y * 512;
    const uint8_t* rowU = fr + ((y - 1) & 511) * 512;
    const uint8_t* rowD = fr + ((y + 1) & 511) * 512;

    int mn0 = IMAX, mn1 = IMAX, mn2 = IMAX, mn3 = IMAX;
    int mx0 = IMIN, mx1 = IMIN, mx2 = IMIN, mx3 = IMIN;

    for (int gx = 0; gx < 64; gx += 4) {
        const int x = x0 + gx;
        const uint32_t c4 = *(const uint32_t*)(rowC + x);
        const uint32_t u4 = *(const uint32_t*)(rowU + x);
        const uint32_t d4 = *(const uint32_t*)(rowD + x);
        const int xm1 = (x - 1) & 511;
        const int cp = rowC[xm1], up = rowU[xm1], dp = rowD[xm1];
#pragma unroll
        for (int k = 0; k < 4; ++k) {
            const int g  = (c4 >> (8 * k)) & 255;
            const int s0 = (u4 >> (8 * k)) & 255;
            const int s1 = k ? (int)((u4 >> (8 * (k - 1))) & 255) : up;
            const int s2 = k ? (int)((c4 >> (8 * (k - 1))) & 255) : cp;
            const int s3 = k ? (int)((d4 >> (8 * (k - 1))) & 255) : dp;
            const int j0 = g * 256 + s0, j1 = g * 256 + s1;
            const int j2 = g * 256 + s2, j3 = g * 256 + s3;
            mn0 = min(mn0, j0); mx0 = max(mx0, j0);
            mn1 = min(mn1, j1); mx1 = max(mx1, j1);
            mn2 = min(mn2, j2); mx2 = max(mx2, j2);
            mn3 = min(mn3, j3); mx3 = max(mx3, j3);
        }
    }

    __shared__ int smn[4], smx[4];
    if (threadIdx.x < 4) { smn[threadIdx.x] = IMAX; smx[threadIdx.x] = IMIN; }
    __syncthreads();
    atomicMin(&smn[0], mn0); atomicMax(&smx[0], mx0);
    atomicMin(&smn[1], mn1); atomicMax(&smx[1], mx1);
    atomicMin(&smn[2], mn2); atomicMax(&smx[2], mx2);
    atomicMin(&smn[3], mn3); atomicMax(&smx[3], mx3);
    __syncthreads();
    if (threadIdx.x < 4) {
        atomicMin(&vm[(img * 4 + threadIdx.x) * 2],     smn[threadIdx.x]);
        atomicMax(&vm[(img * 4 + threadIdx.x) * 2 + 1], smx[threadIdx.x]);
    }
}

// ---------------------------------------------------------------------------
// Faithful histc binning + (i-j)^2 accumulation. Per-thread u32 partials are
// exact (<= 64*65025 < 2^24); per-block u64 via ds_add_u64; block partial is
// split into hi (multiple of 2^16) / lo (<2^16) f32 values, both exact.
__global__ void k_sum(const uint8_t* __restrict__ gray,
                      const int* __restrict__ vm,
                      float* __restrict__ Phi, float* __restrict__ Plo) {
    const int img = blockIdx.y;
    int   vmin_[4];
    float scale_[4];
#pragma unroll
    for (int a = 0; a < 4; ++a) {
        const int vmn = vm[(img * 4 + a) * 2];
        const int vmx = vm[(img * 4 + a) * 2 + 1];
        vmin_[a]  = vmn;
        scale_[a] = 65536.0f / fmaxf((float)(vmx - vmn), 1.0f);
    }

    const int T = blockIdx.x * 256 + threadIdx.x;
    const long P = (long)T * 64;
    const int n  = (int)(P >> 18);
    const int q  = (int)(P & (HWPX - 1));
    const int y  = q >> 9;
    const int x0 = q & 511;
    const uint8_t* fr   = gray + (size_t)img * NPX + (size_t)n * HWPX;
    const uint8_t* rowC = fr + y * 512;
    const uint8_t* rowU = fr + ((y - 1) & 511) * 512;
    const uint8_t* rowD = fr + ((y + 1) & 511) * 512;
    __builtin_prefetch(rowU + x0, 0, 0);
    __builtin_prefetch(rowD + x0, 0, 0);

    uint32_t acc0 = 0, acc1 = 0, acc2 = 0, acc3 = 0;
    for (int gx = 0; gx < 64; gx += 4) {
        const int x = x0 + gx;
        const uint32_t c4 = *(const uint32_t*)(rowC + x);
        const uint32_t u4 = *(const uint32_t*)(rowU + x);
        const uint32_t d4 = *(const uint32_t*)(rowD + x);
        const int xm1 = (x - 1) & 511;
        const int cp = rowC[xm1], up = rowU[xm1], dp = rowD[xm1];
#pragma unroll
        for (int k = 0; k < 4; ++k) {
            const int g  = (c4 >> (8 * k)) & 255;
            const int s0 = (u4 >> (8 * k)) & 255;
            const int s1 = k ? (int)((u4 >> (8 * (k - 1))) & 255) : up;
            const int s2 = k ? (int)((c4 >> (8 * (k - 1))) & 255) : cp;
            const int s3 = k ? (int)((d4 >> (8 * (k - 1))) & 255) : dp;
            const int jj[4] = { g * 256 + s0, g * 256 + s1, g * 256 + s2, g * 256 + s3 };
            uint32_t* accs[4] = { &acc0, &acc1, &acc2, &acc3 };
#pragma unroll
            for (int a = 0; a < 4; ++a) {
                int b = (int)floorf((float)(jj[a] - vmin_[a]) * scale_[a]);
                b = min(b, 65535);                       // clip (lower bound is 0)
                const int d = (b >> 8) - (b & 255);
                *accs[a] += (uint32_t)(d * d);
            }
        }
    }

    __shared__ unsigned long long ssum[4];
    if (threadIdx.x < 4) ssum[threadIdx.x] = 0ull;
    __syncthreads();
    atomicAdd(&ssum[0], (unsigned long long)acc0);
    atomicAdd(&ssum[1], (unsigned long long)acc1);
    atomicAdd(&ssum[2], (unsigned long long)acc2);
    atomicAdd(&ssum[3], (unsigned long long)acc3);
    __syncthreads();
    if (threadIdx.x == 0) {
#pragma unroll
        for (int a = 0; a < 4; ++a) {
            const unsigned long long s = ssum[a];
            const unsigned long long lo = s & 0xFFFFull;
            const int idx = (img * 4 + a) * 256 + blockIdx.x;
            Phi[idx] = (float)(s - lo);                  // multiple of 2^16, exact
            Plo[idx] = (float)lo;                        // < 2^16, exact
        }
    }
}

// ---------------------------------------------------------------------------
// Final reduction: one wave (EXEC all-1s). For each of 8 (image,angle) combos
// and each part (hi/lo), chain 4x v_wmma_f32_16x16x4_f32 with B = ones:
//   D = A*1 + C  => D[m][n] = running row-sums (independent of n).
// A layout (documented): lane l<16 -> {A[l][0],A[l][1]}, l>=16 -> {A[l-16][2],A[l-16][3]};
// we just need each of the 64 partials in a distinct A slot. Total is
// lane0(d0..d7) + lane16(d0..d7) per the documented C/D layout. All sums are
// exact in f32 by construction; combine hi/lo in double.
__global__ void k_final(const float* __restrict__ Phi,
                        const float* __restrict__ Plo,
                        float* __restrict__ out) {
    __shared__ float red[32];
    const int l = threadIdx.x;
    v2f ones; ones.x = 1.0f; ones.y = 1.0f;

    double tot[8];
    for (int c = 0; c < 8; ++c) {
        double tc = 0.0;
        for (int part = 0; part < 2; ++part) {
            const float* __restrict__ Pp = part ? Plo : Phi;
            v8f acc = {0.f, 0.f, 0.f, 0.f, 0.f, 0.f, 0.f, 0.f};
#pragma unroll
            for (int w = 0; w < 4; ++w) {
                const float* bp = Pp + c * 256 + w * 64;
                v2f a; a.x = bp[2 * l]; a.y = bp[2 * l + 1];
                acc = __builtin_amdgcn_wmma_f32_16x16x4_f32(
                    false, a, false, ones, (short)0, acc, false, false);
            }
            const float s = acc[0] + acc[1] + acc[2] + acc[3] +
                            acc[4] + acc[5] + acc[6] + acc[7];
            red[l] = s;
            __syncthreads();
            const float t = red[0] + red[16];
            __syncthreads();
            tc += (double)t;
        }
        tot[c] = tc;
    }

    if (l == 0) {
        const double inv = 1.0 / (4.0 * (double)NPX);    // glcm normalization
        double loss = 0.0;
        for (int a = 0; a < 4; ++a)
            loss += fabs(tot[a] * inv - tot[4 + a] * inv);
        out[0] = (float)(loss * 0.25);                   // mean over 4 angles
    }
}

// ---------------------------------------------------------------------------
extern "C" void kernel_launch(void* const* d_in, const int* in_sizes, int n_in,
                              void* d_out, int out_size, void* d_ws, size_t ws_size,
                              hipStream_t stream) {
    (void)in_sizes; (void)n_in; (void)out_size; (void)ws_size;
    const float* org = (const float*)d_in[0];
    const float* gen = (const float*)d_in[1];
    uint8_t* gray = (uint8_t*)d_ws;
    int*   vm  = (int*)((char*)d_ws + VM_OFF);
    float* Phi = (float*)((char*)d_ws + PHI_OFF);
    float* Plo = (float*)((char*)d_ws + PLO_OFF);

    k_init  <<<dim3(1),        32,  0, stream>>>(vm);
    k_gray  <<<dim3(4096, 2),  256, 0, stream>>>(org, gen, gray);
    k_minmax<<<dim3(256, 2),   256, 0, stream>>>(gray, vm);
    k_sum   <<<dim3(256, 2),   256, 0, stream>>>(gray, vm, Phi, Plo);
    k_final <<<dim3(1),        32,  0, stream>>>(Phi, Plo, (float*)d_out);
}